// HeteroRGCNLayer_46445776339627
// MI455X (gfx1250) — compile-verified
//
#include <hip/hip_runtime.h>

typedef __attribute__((ext_vector_type(2))) float v2f;
typedef __attribute__((ext_vector_type(8))) float v8f;

#define DFEAT 128
#define TILE  16

// ---------------------------------------------------------------------------
// Zero-fill (h accumulator must be zeroed every call; harness poisons ws).
// ---------------------------------------------------------------------------
__global__ void zero_f4(float4* __restrict__ p, long long n4) {
  long long i = (long long)blockIdx.x * blockDim.x + threadIdx.x;
  if (i < n4) p[i] = make_float4(0.f, 0.f, 0.f, 0.f);
}

// ---------------------------------------------------------------------------
// WhC = feats @ Wc^T + bc  AND  WhE = feats @ We^T + be, fused: one wave32
// computes one 16x16 tile of BOTH outputs, sharing the A fragments (halves
// feats traffic). K stepped by 4 via V_WMMA_F32_16X16X4_F32 (exact f32).
//
// A fragment (16x4, MxK): lane&15 = M, VGPR v -> K = (lane>>4)*2 + v  (float2)
// B fragment (4x16, KxN): B[k][n] = W[n][k]; lane&15 = N, same K split ->
//                         contiguous float2 from row-major W.
// C/D (16x16): VGPR r -> M = r + (lane>>4)*8, N = lane&15.
// EXEC stays all-ones for WMMA: A-row loads clamped; only tail-tile stores
// are per-lane predicated (behind a wave-uniform branch).
// ---------------------------------------------------------------------------
__global__ __launch_bounds__(256) void rgcn_proj_gemm(
    const float* __restrict__ feats,
    const float* __restrict__ Wc, const float* __restrict__ bc,
    const float* __restrict__ We, const float* __restrict__ be,
    float* __restrict__ WhC, float* __restrict__ WhE, int n)
{
  const int wave  = threadIdx.x >> 5;
  const int lane  = threadIdx.x & 31;
  const int gw    = blockIdx.x * 8 + wave;
  const int nColT = DFEAT / TILE;                  // 8 col tiles
  const int m0    = (gw / nColT) * TILE;
  const int n0    = (gw % nColT) * TILE;
  if (m0 >= n) return;                             // wave-uniform

  const int half = lane >> 4;                      // 0 | 1
  const int l15  = lane & 15;

  int arowi = m0 + l15;
  if (arowi >= n) arowi = n - 1;                   // clamp: keep EXEC all-1s
  const float* arow  = feats + (size_t)arowi * DFEAT + half * 2;
  const float* bcrow = Wc    + (size_t)(n0 + l15) * DFEAT + half * 2;
  const float* berow = We    + (size_t)(n0 + l15) * DFEAT + half * 2;

  v8f accC = {0.f, 0.f, 0.f, 0.f, 0.f, 0.f, 0.f, 0.f};
  v8f accE = {0.f, 0.f, 0.f, 0.f, 0.f, 0.f, 0.f, 0.f};
#pragma unroll 8
  for (int k = 0; k < DFEAT; k += 4) {
    v2f a   = *(const v2f*)(arow + k);
    v2f bC  = *(const v2f*)(bcrow + k);
    v2f bE  = *(const v2f*)(berow + k);
    accC = __builtin_amdgcn_wmma_f32_16x16x4_f32(
        false, a, false, bC, (short)0, accC, false, false);
    accE = __builtin_amdgcn_wmma_f32_16x16x4_f32(
        false, a, false, bE, (short)0, accE, false, false);
  }

  const float biasC = bc[n0 + l15];
  const float biasE = be[n0 + l15];
  float* ocol = WhC + (size_t)(m0 + half * 8) * DFEAT + n0 + l15;
  float* ecol = WhE + (size_t)(m0 + half * 8) * DFEAT + n0 + l15;

  if (m0 + TILE <= n) {                            // wave-uniform fast path
#pragma unroll
    for (int r = 0; r < 8; ++r) {
      ocol[(size_t)r * DFEAT] = accC[r] + biasC;
      ecol[(size_t)r * DFEAT] = accE[r] + biasE;
    }
  } else {                                         // tail tile (n % 16 != 0)
#pragma unroll
    for (int r = 0; r < 8; ++r) {
      const int M = m0 + half * 8 + r;
      if (M < n) {
        ocol[(size_t)r * DFEAT] = accC[r] + biasC;
        ecol[(size_t)r * DFEAT] = accE[r] + biasE;
      }
    }
  }
}

// ---------------------------------------------------------------------------
// out = h @ Wo^T + bo + WhE   (fused epilogue). Same WMMA tiling as above.
// ---------------------------------------------------------------------------
__global__ __launch_bounds__(256) void rgcn_out_gemm(
    const float* __restrict__ h,
    const float* __restrict__ Wo, const float* __restrict__ bo,
    const float* __restrict__ WhE, float* __restrict__ out, int n)
{
  const int wave  = threadIdx.x >> 5;
  const int lane  = threadIdx.x & 31;
  const int gw    = blockIdx.x * 8 + wave;
  const int nColT = DFEAT / TILE;
  const int m0    = (gw / nColT) * TILE;
  const int n0    = (gw % nColT) * TILE;
  if (m0 >= n) return;

  const int half = lane >> 4;
  const int l15  = lane & 15;

  int arowi = m0 + l15;
  if (arowi >= n) arowi = n - 1;
  const float* arow = h  + (size_t)arowi * DFEAT + half * 2;
  const float* brow = Wo + (size_t)(n0 + l15) * DFEAT + half * 2;

  v8f acc = {0.f, 0.f, 0.f, 0.f, 0.f, 0.f, 0.f, 0.f};
#pragma unroll 8
  for (int k = 0; k < DFEAT; k += 4) {
    v2f a = *(const v2f*)(arow + k);
    v2f b = *(const v2f*)(brow + k);
    acc = __builtin_amdgcn_wmma_f32_16x16x4_f32(
        false, a, false, b, (short)0, acc, false, false);
  }

  const float bias = bo[n0 + l15];
  const size_t base = (size_t)(m0 + half * 8) * DFEAT + n0 + l15;

  if (m0 + TILE <= n) {                            // wave-uniform fast path
#pragma unroll
    for (int r = 0; r < 8; ++r) {
      const size_t idx = base + (size_t)r * DFEAT;
      out[idx] = acc[r] + bias + WhE[idx];
    }
  } else {
#pragma unroll
    for (int r = 0; r < 8; ++r) {
      const int M = m0 + half * 8 + r;
      if (M < n) {
        const size_t idx = base + (size_t)r * DFEAT;
        out[idx] = acc[r] + bias + WhE[idx];
      }
    }
  }
}

// ---------------------------------------------------------------------------
// h[dst[e], :] += Wh[src[e], :]  -- 128 threads per edge, f32 global atomics.
// This is the bandwidth-bound phase (~1 GB of traffic across both etypes,
// ~43 us at 23.3 TB/s peak); unsorted segment-sum => global_atomic_add_f32.
// ---------------------------------------------------------------------------
__global__ __launch_bounds__(256) void scatter_add_edges(
    const float* __restrict__ Wh,
    const int* __restrict__ src, const int* __restrict__ dst,
    float* __restrict__ h, int nE)
{
  const long long idx = (long long)blockIdx.x * blockDim.x + threadIdx.x;
  const int e = (int)(idx >> 7);
  if (e >= nE) return;
  const int d = (int)(idx & 127);
  const int s = src[e];
  const int t = dst[e];
  atomicAdd(h + (size_t)t * DFEAT + d, Wh[(size_t)s * DFEAT + d]);
}

// ---------------------------------------------------------------------------
extern "C" void kernel_launch(void* const* d_in, const int* in_sizes, int n_in,
                              void* d_out, int out_size, void* d_ws, size_t ws_size,
                              hipStream_t stream) {
  const float* feats = (const float*)d_in[0];
  const float* Wc    = (const float*)d_in[1];
  const float* bc    = (const float*)d_in[2];
  const float* We    = (const float*)d_in[3];
  const float* be    = (const float*)d_in[4];
  const float* Wo    = (const float*)d_in[5];
  const float* bo    = (const float*)d_in[6];
  const int*   srcC  = (const int*)d_in[7];
  const int*   dstC  = (const int*)d_in[8];
  const int*   srcE  = (const int*)d_in[9];
  const int*   dstE  = (const int*)d_in[10];

  const int n   = in_sizes[0] / DFEAT;   // 50000 nodes
  const int nEc = in_sizes[7];           // 500000 chem edges
  const int nEe = in_sizes[9];           // 500000 elec edges

  const size_t feSize = (size_t)n * DFEAT;   // floats per [N,D] buffer
  float* WhC = (float*)d_ws;
  float* WhE = WhC + feSize;
  float* h   = WhE + feSize;
  float* out = (float*)d_out;

  // 1) zero the scatter accumulator
  const long long n4 = (long long)feSize / 4;
  zero_f4<<<(unsigned)((n4 + 255) / 256), 256, 0, stream>>>((float4*)h, n4);

  // 2) fused per-etype projections (8 waves/block, 1 dual tile/wave)
  const int tiles  = ((n + TILE - 1) / TILE) * (DFEAT / TILE);  // 25000
  const int blocks = (tiles + 7) / 8;                            // 3125
  rgcn_proj_gemm<<<dim3(blocks, 1), 256, 0, stream>>>(
      feats, Wc, bc, We, be, WhC, WhE, n);

  // 3) scatter-add per etype
  const long long workC = (long long)nEc * DFEAT;
  scatter_add_edges<<<(unsigned)((workC + 255) / 256), 256, 0, stream>>>(
      WhC, srcC, dstC, h, nEc);
  const long long workE = (long long)nEe * DFEAT;
  scatter_add_edges<<<(unsigned)((workE + 255) / 256), 256, 0, stream>>>(
      WhE, srcE, dstE, h, nEe);

  // 4) output GEMM with fused (+ b_out + Wh_elec)
  rgcn_out_gemm<<<dim3(blocks, 1), 256, 0, stream>>>(h, Wo, bo, WhE, out, n);
}